// Net_17205638988409
// MI455X (gfx1250) — compile-verified
//
#include <hip/hip_runtime.h>

typedef __attribute__((ext_vector_type(2))) float v2f;
typedef __attribute__((ext_vector_type(8))) float v8f;

#define FD 64   // feature dim D == F == 64

static __device__ __forceinline__ v8f wmma4(v2f a, v2f b, v8f c) {
  // D = A(16x4,f32) * B(4x16,f32) + C(16x16,f32)
  return __builtin_amdgcn_wmma_f32_16x16x4_f32(false, a, false, b, (short)0, c,
                                               false, false);
}

// ----------------- elementwise helpers -----------------
__global__ void zero_kernel(float* __restrict__ p, int n) {
  int i = blockIdx.x * blockDim.x + threadIdx.x;
  if (i < n) p[i] = 0.0f;
}

__global__ void copy_kernel(float* __restrict__ d, const float* __restrict__ s, int n) {
  int i = blockIdx.x * blockDim.x + threadIdx.x;
  if (i < n) d[i] = s[i];
}

// ----------------- edge scatter: agg[dst] += h[src] -----------------
// 16 threads per edge, each moves 4 consecutive floats (float4 gather, 4 atomics)
__global__ void scatter_kernel(float* __restrict__ agg, const float* __restrict__ h,
                               const int* __restrict__ src, const int* __restrict__ dst,
                               int E) {
  long long t = (long long)blockIdx.x * blockDim.x + threadIdx.x;
  int e = (int)(t >> 4);
  if (e >= E) return;
  int q = (int)(t & 15);
  int s = src[e], d = dst[e];
  const float4 v = reinterpret_cast<const float4*>(h)[(long long)s * (FD / 4) + q];
  float* a = agg + (long long)d * FD + q * 4;
  atomicAdd(a + 0, v.x);
  atomicAdd(a + 1, v.y);
  atomicAdd(a + 2, v.z);
  atomicAdd(a + 3, v.w);
}

// ----------------- per-node MLP: z = relu(relu(zin@wa+ba)@wb+bb) + BN stats ----
// block = 128 threads = 4 waves; block handles a 16-row tile; wave w owns cols [16w,16w+16)
__global__ void __launch_bounds__(128)
mlp_kernel(float* __restrict__ zout, const float* __restrict__ zin,
           const float* __restrict__ wa, const float* __restrict__ ba,
           const float* __restrict__ wb, const float* __restrict__ bb,
           float* __restrict__ bnsum, float* __restrict__ bnsq, int nrows) {
  __shared__ float tls[16][72];  // staged intermediate t (16 rows x 64 cols, padded)

  const int lane = threadIdx.x & 31;
  const int wave = threadIdx.x >> 5;
  const int col  = lane & 15;
  const int half = lane >> 4;
  const int n0   = wave * 16;
  const int row0 = blockIdx.x * 16;

  // ---- preload weight fragments into registers (B layout: N=col, K=4k0+2h+{0,1}) ----
  v2f wafr[16], wbfr[16];
#pragma unroll
  for (int k0 = 0; k0 < 16; ++k0) {
    const int k = 4 * k0 + 2 * half;
    wafr[k0].x = wa[k * FD + n0 + col];
    wafr[k0].y = wa[(k + 1) * FD + n0 + col];
    wbfr[k0].x = wb[k * FD + n0 + col];
    wbfr[k0].y = wb[(k + 1) * FD + n0 + col];
  }

  // ---- GEMM1: t = relu(zin_tile @ wa + ba) ----
  const int arow = min(row0 + col, nrows - 1);  // uniform-safe clamp (N%16==0 -> no-op)
  const v2f* zin2 = reinterpret_cast<const v2f*>(zin) + (long long)arow * (FD / 2);
  v8f acc;
  {
    const float bias = ba[n0 + col];
#pragma unroll
    for (int v = 0; v < 8; ++v) acc[v] = bias;
  }
#pragma unroll
  for (int k0 = 0; k0 < 16; ++k0) {
    v2f a = zin2[2 * k0 + half];  // A layout: M=col, K=4k0+2h+{0,1} (contiguous pair)
    acc = wmma4(a, wafr[k0], acc);
  }
#pragma unroll
  for (int v = 0; v < 8; ++v) acc[v] = fmaxf(acc[v], 0.0f);

  // ---- stage t through LDS (each wave writes its own 16-col stripe) ----
#pragma unroll
  for (int v = 0; v < 8; ++v) tls[v + 8 * half][n0 + col] = acc[v];
  __syncthreads();

  // ---- GEMM2: z = relu(t @ wb + bb) ----
  {
    const float bias = bb[n0 + col];
#pragma unroll
    for (int v = 0; v < 8; ++v) acc[v] = bias;
  }
#pragma unroll
  for (int k0 = 0; k0 < 16; ++k0) {
    const int k = 4 * k0 + 2 * half;
    v2f a;
    a.x = tls[col][k];
    a.y = tls[col][k + 1];
    acc = wmma4(a, wbfr[k0], acc);
  }

  // ---- ReLU, store, fused BatchNorm partial sums ----
  float s = 0.0f, sq = 0.0f;
#pragma unroll
  for (int v = 0; v < 8; ++v) {
    const int r = row0 + v + 8 * half;
    const float z = fmaxf(acc[v], 0.0f);
    if (r < nrows) {
      zout[(long long)r * FD + n0 + col] = z;
      s += z;
      sq += z * z;
    }
  }
  atomicAdd(&bnsum[n0 + col], s);
  atomicAdd(&bnsq[n0 + col], sq);
}

// ----------------- BatchNorm apply; also re-seeds next layer's agg -----------------
__global__ void bnapply_kernel(float* __restrict__ hout, float* __restrict__ aggout,
                               const float* __restrict__ z,
                               const float* __restrict__ bnsum, const float* __restrict__ bnsq,
                               const float* __restrict__ g, const float* __restrict__ be,
                               int nrows) {
  long long t = (long long)blockIdx.x * blockDim.x + threadIdx.x;
  if (t >= (long long)nrows * FD) return;
  const int f = (int)(t & (FD - 1));
  const float invN = 1.0f / (float)nrows;
  const float mean = bnsum[f] * invN;
  const float var  = bnsq[f] * invN - mean * mean;
  const float o = (z[t] - mean) * rsqrtf(var + 1e-5f) * g[f] + be[f];
  hout[t] = o;
  if (aggout) aggout[t] = o;
}

// ----------------- global_add_pool: xpool[batch[i]] += h[i] -----------------
__global__ void pool_kernel(float* __restrict__ xpool, const float* __restrict__ h,
                            const int* __restrict__ batch, int nrows) {
  long long t = (long long)blockIdx.x * blockDim.x + threadIdx.x;
  int i = (int)(t >> 4);
  if (i >= nrows) return;
  int q = (int)(t & 15);
  int gidx = batch[i];
  const float4 v = reinterpret_cast<const float4*>(h)[(long long)i * (FD / 4) + q];
  float* p = xpool + (long long)gidx * FD + q * 4;
  atomicAdd(p + 0, v.x);
  atomicAdd(p + 1, v.y);
  atomicAdd(p + 2, v.z);
  atomicAdd(p + 3, v.w);
}

// ----------------- head: out = relu((xpool@w0+b0)@w1+b1) -----------------
// block = 4 waves; each wave owns one 16-row tile; t[16][128] staged per-wave in LDS
__global__ void __launch_bounds__(128)
head_kernel(float* __restrict__ out, const float* __restrict__ xpool,
            const float* __restrict__ w0, const float* __restrict__ b0,
            const float* __restrict__ w1, const float* __restrict__ b1, int G) {
  __shared__ float tls[4][16][132];
  const int lane = threadIdx.x & 31;
  const int wave = threadIdx.x >> 5;
  const int col  = lane & 15;
  const int half = lane >> 4;
  int row0 = (blockIdx.x * 4 + wave) * 16;
  if (row0 + 16 > G) row0 = (G > 16) ? (G - 16) : 0;  // uniform clamp; dup writes benign

  const v2f* x2 = reinterpret_cast<const v2f*>(xpool) + (long long)(row0 + col) * (FD / 2);

  // GEMM1: t = xpool_tile @ w0 + b0   (K = 64, N = 128; no relu)
  for (int nt = 0; nt < 8; ++nt) {
    const int n0 = nt * 16;
    v8f acc;
    const float bias = b0[n0 + col];
#pragma unroll
    for (int v = 0; v < 8; ++v) acc[v] = bias;
#pragma unroll
    for (int k0 = 0; k0 < 16; ++k0) {
      const int k = 4 * k0 + 2 * half;
      v2f a = x2[2 * k0 + half];
      v2f b;
      b.x = w0[k * 128 + n0 + col];
      b.y = w0[(k + 1) * 128 + n0 + col];
      acc = wmma4(a, b, acc);
    }
#pragma unroll
    for (int v = 0; v < 8; ++v) tls[wave][v + 8 * half][n0 + col] = acc[v];
  }
  __syncthreads();

  // GEMM2: out = relu(t @ w1 + b1)   (K = 128, N = 64)
  for (int nt = 0; nt < 4; ++nt) {
    const int n0 = nt * 16;
    v8f acc;
    const float bias = b1[n0 + col];
#pragma unroll
    for (int v = 0; v < 8; ++v) acc[v] = bias;
#pragma unroll
    for (int k0 = 0; k0 < 32; ++k0) {
      const int k = 4 * k0 + 2 * half;
      v2f a;
      a.x = tls[wave][col][k];
      a.y = tls[wave][col][k + 1];
      v2f b;
      b.x = w1[k * FD + n0 + col];
      b.y = w1[(k + 1) * FD + n0 + col];
      acc = wmma4(a, b, acc);
    }
#pragma unroll
    for (int v = 0; v < 8; ++v) {
      out[(long long)(row0 + v + 8 * half) * FD + n0 + col] = fmaxf(acc[v], 0.0f);
    }
  }
}

// ----------------- driver -----------------
extern "C" void kernel_launch(void* const* d_in, const int* in_sizes, int n_in,
                              void* d_out, int out_size, void* d_ws, size_t ws_size,
                              hipStream_t stream) {
  const float* x     = (const float*)d_in[0];
  const int*   eidx  = (const int*)d_in[1];
  const int*   batch = (const int*)d_in[2];
  const float* W[22];
  for (int i = 0; i < 22; ++i) W[i] = (const float*)d_in[3 + i];
  // W: [0..5]=layer1 {wa,ba,wb,bb,g,be}, [6..11]=layer2, [12..17]=layer3,
  //    [18]=w_lin0 [19]=b_lin0 [20]=w_fc1 [21]=b_fc1

  const int N = in_sizes[0] / FD;
  const int E = in_sizes[1] / 2;
  const int G = out_size / FD;

  float* A   = (float*)d_ws;                 // agg buffer [N][64]
  float* Z   = A + (size_t)N * FD;           // pre-BN activations
  float* H   = Z + (size_t)N * FD;           // normalized activations
  float* XP  = H + (size_t)N * FD;           // pooled [G][64]
  float* SUM = XP + (size_t)G * FD;          // BN sum [64]
  float* SQ  = SUM + FD;                     // BN sumsq [64]
  float* out = (float*)d_out;

  const int* src = eidx;
  const int* dst = eidx + E;

  const int  nvec       = N * FD;
  const long long sc_t  = (long long)E * 16;
  const long long pl_t  = (long long)N * 16;
  const int  mlp_blocks = (N + 15) / 16;
  auto blocks = [](long long n) { return dim3((unsigned)((n + 255) / 256)); };

  const float* hin = x;
  for (int layer = 0; layer < 3; ++layer) {
    const float* wa = W[layer * 6 + 0];
    const float* ba = W[layer * 6 + 1];
    const float* wb = W[layer * 6 + 2];
    const float* bb = W[layer * 6 + 3];
    const float* g  = W[layer * 6 + 4];
    const float* be = W[layer * 6 + 5];

    if (layer == 0) copy_kernel<<<blocks(nvec), 256, 0, stream>>>(A, x, nvec);
    zero_kernel<<<1, 2 * FD, 0, stream>>>(SUM, 2 * FD);  // zeros SUM and SQ
    scatter_kernel<<<blocks(sc_t), 256, 0, stream>>>(A, hin, src, dst, E);
    mlp_kernel<<<mlp_blocks, 128, 0, stream>>>(Z, A, wa, ba, wb, bb, SUM, SQ, N);
    bnapply_kernel<<<blocks(nvec), 256, 0, stream>>>(
        H, (layer < 2) ? A : (float*)nullptr, Z, SUM, SQ, g, be, N);
    hin = H;
  }

  zero_kernel<<<blocks((long long)G * FD), 256, 0, stream>>>(XP, G * FD);
  pool_kernel<<<blocks(pl_t), 256, 0, stream>>>(XP, H, batch, N);
  head_kernel<<<(G + 63) / 64, 128, 0, stream>>>(out, XP, W[18], W[19], W[20], W[21], G);
}